// Top2Gate_51900384805184
// MI455X (gfx1250) — compile-verified
//
#include <hip/hip_runtime.h>
#include <hip/hip_bf16.h>
#include <math.h>

// Problem constants (from reference): G=4, S=2048, M=1024, E=16, C=256
#define GQ 4
#define SQ 2048
#define MQ 1024
#define EQ 16
#define CQ 256
#define NTOK (GQ * SQ)            // 8192
#define MIN_LOGIT (-3.402823466e38f)

typedef float v2f __attribute__((ext_vector_type(2)));
typedef float v8f __attribute__((ext_vector_type(8)));

// ---------------------------------------------------------------------------
// Prolog: transpose wg [M,E] -> wgT [E,M] so B-operand fetches per lane are
// two consecutive floats (one global_load_b64) instead of two strided b32s.
// ---------------------------------------------------------------------------
__global__ __launch_bounds__(256)
void transpose_wg_kernel(const float* __restrict__ wg, float* __restrict__ wgT)
{
    const int i = blockIdx.x * 256 + threadIdx.x;   // 0 .. E*M-1
    const int e = i >> 10;                          // wgT row
    const int m = i & (MQ - 1);
    wgT[i] = wg[m * EQ + e];
}

// ---------------------------------------------------------------------------
// Stage 1: logits = x @ wg via V_WMMA_F32_16X16X4_F32, software-pipelined with
// 8 register slots (prefetch depth = 8 K-steps), fused with softmax,
// argmax1/argmax2 (+gumbel), per-token gate values, and me accumulation.
// Grid: 64 blocks x 256 threads (8 waves). Each wave: one 16-row tile x E=16.
// ---------------------------------------------------------------------------
__global__ __launch_bounds__(256)
void gate_logits_kernel(const float* __restrict__ x,
                        const float* __restrict__ wgT,   // [E, M]
                        const float* __restrict__ gumbel,
                        int* __restrict__ idx1, int* __restrict__ idx2,
                        float* __restrict__ gate1, float* __restrict__ gate2,
                        float* __restrict__ me_sum)
{
    __shared__ float tile[8][16][16];   // per-wave 16x16 logits tile
    __shared__ float meLDS[EQ];

    const int tid  = threadIdx.x;
    const int w    = tid >> 5;          // wave id in block
    const int lane = tid & 31;
    const int half = lane >> 4;         // 0: lanes 0-15, 1: lanes 16-31
    const int lrow = lane & 15;
    const int rowBase = blockIdx.x * 128 + w * 16;   // flat token row (g*S+s)
    const int g = blockIdx.x >> 4;      // 16 blocks per group

    if (tid < EQ) meLDS[tid] = 0.0f;
    __syncthreads();

    // ---- WMMA GEMM: C[16x16] += A[16x4] * B[4x16] over K=1024 ----
    // A lane layout: row = lrow, a[j] = x[row][k + 2*half + j]
    // B lane layout: b[j] = B[k + 2*half + j][lrow] = wgT[lrow][k + 2*half + j]
    const float* xrow = x   + (size_t)(rowBase + lrow) * MQ + 2 * half;
    const float* brow = wgT + (size_t)lrow * MQ + 2 * half;

    constexpr int PF = 8;               // prefetch depth (K-steps in flight)
    v2f a[PF], b[PF];
#pragma unroll
    for (int i = 0; i < PF; ++i) {
        a[i] = *(const v2f*)(xrow + 4 * i);
        b[i] = *(const v2f*)(brow + 4 * i);
    }

    v8f c = {};
    for (int kb = 0; kb < MQ - PF * 4; kb += PF * 4) {
#pragma unroll
        for (int i = 0; i < PF; ++i) {
            c = __builtin_amdgcn_wmma_f32_16x16x4_f32(
                    false, a[i], false, b[i], (short)0, c, false, false);
            const int knext = kb + PF * 4 + 4 * i;   // prefetch next block's slot
            a[i] = *(const v2f*)(xrow + knext);
            b[i] = *(const v2f*)(brow + knext);
        }
    }
#pragma unroll
    for (int i = 0; i < PF; ++i) {      // epilogue: drain last PF slots
        c = __builtin_amdgcn_wmma_f32_16x16x4_f32(
                false, a[i], false, b[i], (short)0, c, false, false);
    }

    // C layout: c[r] = logits[rowBase + r + 8*half][lrow] -> transpose via LDS
#pragma unroll
    for (int r = 0; r < 8; ++r)
        tile[w][r + 8 * half][lrow] = c[r];

    // ---- per-row gating: lanes 0-15 each own one token row ----
    if (lane < 16) {
        const int row   = lane;
        const int sflat = rowBase + row;
        float lg[EQ];
#pragma unroll
        for (int e = 0; e < EQ; ++e) lg[e] = tile[w][row][e];

        // argmax of logits == argmax of gates (softmax monotonic); first-max.
        float mx = lg[0]; int i1 = 0;
#pragma unroll
        for (int e = 1; e < EQ; ++e)
            if (lg[e] > mx) { mx = lg[e]; i1 = e; }

        float sum = 0.0f;
#pragma unroll
        for (int e = 0; e < EQ; ++e) sum += expf(lg[e] - mx);
        const float inv = 1.0f / sum;
        const float g1  = inv;           // exp(mx - mx) * inv

        // second argmax over logits + gumbel, with idx1 masked to MIN_LOGIT
        const float* gum = gumbel + (size_t)sflat * EQ;
        float best2 = -INFINITY, lgAt2 = lg[0];
        int   i2 = 0;
#pragma unroll
        for (int e = 0; e < EQ; ++e) {
            float v = (e == i1) ? MIN_LOGIT : (lg[e] + gum[e]);
            if (v > best2) { best2 = v; i2 = e; lgAt2 = lg[e]; }
        }
        const float g2 = expf(lgAt2 - mx) * inv;

        idx1[sflat]  = i1;
        idx2[sflat]  = i2;
        gate1[sflat] = g1;
        gate2[sflat] = g2;

        // me accumulation: sum of gates over tokens, per expert
#pragma unroll
        for (int e = 0; e < EQ; ++e)
            atomicAdd(&meLDS[e], expf(lg[e] - mx) * inv);
    }
    __syncthreads();
    if (tid < EQ) atomicAdd(&me_sum[g * EQ + tid], meLDS[tid]);
}

// ---------------------------------------------------------------------------
// Stage 2: ordered rank (cumsum of one-hot masks) per (g, e).
// One wave per group: lanes 0-15 scan idx1 (and record counts1),
// lanes 16-31 scan idx2 with the counts1 offset folded in.
// ---------------------------------------------------------------------------
__global__ __launch_bounds__(32)
void rank_kernel(const int* __restrict__ idx1, const int* __restrict__ idx2,
                 int* __restrict__ loc1, int* __restrict__ loc2,
                 int* __restrict__ counts1)
{
    const int g    = blockIdx.x;
    const int lane = threadIdx.x;
    __shared__ int c1[EQ];

    if (lane < EQ) {
        const int e = lane;
        const int* p = idx1 + g * SQ;
        int cnt = 0;
        for (int s = 0; s < SQ; ++s)
            if (p[s] == e) loc1[g * SQ + s] = cnt++;
        c1[e] = cnt;
        counts1[g * EQ + e] = cnt;
    }
    __syncthreads();
    if (lane >= EQ) {
        const int e = lane - EQ;
        const int* p = idx2 + g * SQ;
        const int base = c1[e];
        int cnt = 0;
        for (int s = 0; s < SQ; ++s)
            if (p[s] == e) loc2[g * SQ + s] = base + cnt++;
    }
}

// ---------------------------------------------------------------------------
// Stage 3: scatter combine_weights / dispatch_mask into the zeroed output,
// and compute l_aux (thread 0).
// Output: out[0] = l_aux; out[1..] = cw [G,S,E,C]; then dm [G,S,E,C].
// ---------------------------------------------------------------------------
__global__ __launch_bounds__(256)
void scatter_kernel(const int* __restrict__ idx1, const int* __restrict__ idx2,
                    const float* __restrict__ gate1, const float* __restrict__ gate2,
                    const int* __restrict__ loc1, const int* __restrict__ loc2,
                    const float* __restrict__ me_sum, const int* __restrict__ counts1,
                    float* __restrict__ out)
{
    const int t = blockIdx.x * blockDim.x + threadIdx.x;   // token id
    if (t == 0) {
        // l_aux = mean_{g,e} (me_sum/S) * (counts1/S)
        float acc = 0.0f;
        for (int i = 0; i < GQ * EQ; ++i)
            acc += (me_sum[i] * (1.0f / SQ)) * ((float)counts1[i] * (1.0f / SQ));
        out[0] = acc * (1.0f / (GQ * EQ));
    }
    if (t >= NTOK) return;

    float* cw = out + 1;
    float* dm = out + 1 + (size_t)NTOK * EQ * CQ;

    const int   i1 = idx1[t], i2 = idx2[t];
    const int   l1 = loc1[t], l2 = loc2[t];
    const bool  k1 = (l1 < CQ), k2 = (l2 < CQ);
    float a = k1 ? gate1[t] : 0.0f;
    float b = k2 ? gate2[t] : 0.0f;
    float d = a + b;
    d = (d > 0.0f) ? d : 1.0f;
    a /= d;
    b /= d;

    const size_t base = (size_t)t * EQ * CQ;
    if (k1) {
        const size_t o = base + (size_t)i1 * CQ + l1;
        cw[o] = a;
        dm[o] = 1.0f;
    }
    if (k2) {
        const size_t o = base + (size_t)i2 * CQ + l2;
        cw[o] = b;
        dm[o] = 1.0f;
    }
}

// ---------------------------------------------------------------------------
extern "C" void kernel_launch(void* const* d_in, const int* in_sizes, int n_in,
                              void* d_out, int out_size, void* d_ws, size_t ws_size,
                              hipStream_t stream)
{
    const float* x   = (const float*)d_in[0];   // [G,S,M] f32
    const float* wg  = (const float*)d_in[1];   // [M,E]   f32
    const float* gum = (const float*)d_in[2];   // [G,S,E] f32
    float* out = (float*)d_out;

    // workspace layout (all 16B-aligned)
    char* ws = (char*)d_ws;
    float* wgT     = (float*)(ws);              // 64 KB  [E, M]
    int*   idx1    = (int*)(ws + 65536);
    int*   idx2    = (int*)(ws + 98304);
    float* gate1   = (float*)(ws + 131072);
    float* gate2   = (float*)(ws + 163840);
    int*   loc1    = (int*)(ws + 196608);
    int*   loc2    = (int*)(ws + 229376);
    float* me_sum  = (float*)(ws + 262144);
    int*   counts1 = (int*)(ws + 262144 + 256);

    // Output is ~268 MB of mostly zeros: zero-fill, then scatter nonzeros.
    hipMemsetAsync(d_out, 0, (size_t)out_size * sizeof(float), stream);
    hipMemsetAsync(me_sum, 0, GQ * EQ * sizeof(float), stream);

    transpose_wg_kernel<<<(EQ * MQ) / 256, 256, 0, stream>>>(wg, wgT);
    gate_logits_kernel<<<64, 256, 0, stream>>>(x, wgT, gum, idx1, idx2,
                                               gate1, gate2, me_sum);
    rank_kernel<<<GQ, 32, 0, stream>>>(idx1, idx2, loc1, loc2, counts1);
    scatter_kernel<<<NTOK / 256, 256, 0, stream>>>(idx1, idx2, gate1, gate2,
                                                   loc1, loc2, me_sum, counts1, out);
}